// PointerGenerator_72164040508034
// MI455X (gfx1250) — compile-verified
//
#include <hip/hip_runtime.h>
#include <math.h>

// Problem constants (static shapes from the reference)
#define B_   64
#define H_   1024
#define E_   512
#define V_   50000
#define S_   400
#define NOOV_ 100
#define EXT_ (V_ + NOOV_)

typedef __attribute__((ext_vector_type(2))) float v2f;
typedef __attribute__((ext_vector_type(8))) float v8f;

__device__ __forceinline__ v8f wmma_f32_k4(v2f a, v2f b, v8f c) {
    // D = A(16x4,f32) x B(4x16,f32) + C(16x16,f32)
    return __builtin_amdgcn_wmma_f32_16x16x4_f32(
        /*neg_a=*/false, a, /*neg_b=*/false, b,
        /*c_mod=*/(short)0, c, /*reuse_a=*/false, /*reuse_b=*/false);
}

// ---------------------------------------------------------------------------
// Kernel 1: p_gen = sigmoid([dec;ctx;emb] . W_pgen + b_pgen), one block per row
// ---------------------------------------------------------------------------
__global__ void __launch_bounds__(256)
pgen_kernel(const float* __restrict__ dec, const float* __restrict__ ctx,
            const float* __restrict__ emb, const float* __restrict__ Wp,
            const float* __restrict__ bp, float* __restrict__ pg_ws,
            float* __restrict__ pg_out) {
    __shared__ float red[256];
    const int b = blockIdx.x, t = threadIdx.x;
    float s = 0.f;
    for (int i = t; i < 2 * H_ + E_; i += 256) {
        float x;
        if (i < H_)            x = dec[b * H_ + i];
        else if (i < 2 * H_)   x = ctx[b * H_ + i - H_];
        else                   x = emb[b * E_ + i - 2 * H_];
        s += x * Wp[i];
    }
    red[t] = s; __syncthreads();
    for (int off = 128; off > 0; off >>= 1) {
        if (t < off) red[t] += red[t + off];
        __syncthreads();
    }
    if (t == 0) {
        const float z = red[0] + bp[0];
        const float p = 1.f / (1.f + expf(-z));
        pg_ws[b]  = p;
        pg_out[b] = p;
    }
}

// ---------------------------------------------------------------------------
// Kernel 2a: partial GEMM for mid: part[kc] = [dec;ctx]_chunk @ W_mid^T_chunk
// K (2048) split into 4 chunks of 512 for occupancy (32 blocks, 256 waves).
// Each chunk lies wholly inside dec (kc=0,1) or ctx (kc=2,3): uniform pointer.
// ---------------------------------------------------------------------------
__global__ void __launch_bounds__(256)
gemm_mid_partial_kernel(const float* __restrict__ dec, const float* __restrict__ ctx,
                        const float* __restrict__ Wm, float* __restrict__ part) {
    const int lane  = threadIdx.x & 31;
    const int wave  = threadIdx.x >> 5;
    const int kc    = blockIdx.x >> 3;            // K chunk 0..3
    const int nblk  = blockIdx.x & 7;
    const int tileN = nblk * 8 + wave;            // 64 N tiles exactly
    const int nloc  = lane & 15;
    const int n     = tileN * 16 + nloc;
    const int kh    = (lane >> 4) << 1;           // 0 or 2

    const int kstart = kc * 512;
    const float* asrc = (kstart < H_) ? dec : ctx;
    const int    abase = (kstart < H_) ? kstart : kstart - H_;

    v8f acc0 = {}, acc1 = {}, acc2 = {}, acc3 = {};
    const float* bp2 = Wm + (size_t)n * (2 * H_) + kstart + kh;  // K contiguous
    const float* ap2 = asrc + (size_t)nloc * H_ + abase + kh;

    for (int k0 = 0; k0 < 512; k0 += 16) {
#pragma unroll
        for (int u = 0; u < 16; u += 4) {
            const v2f bf = *(const v2f*)(bp2 + u);
            const v2f a0 = *(const v2f*)(ap2 + u);
            const v2f a1 = *(const v2f*)(ap2 + u + 16 * H_);
            const v2f a2 = *(const v2f*)(ap2 + u + 32 * H_);
            const v2f a3 = *(const v2f*)(ap2 + u + 48 * H_);
            acc0 = wmma_f32_k4(a0, bf, acc0);
            acc1 = wmma_f32_k4(a1, bf, acc1);
            acc2 = wmma_f32_k4(a2, bf, acc2);
            acc3 = wmma_f32_k4(a3, bf, acc3);
        }
        bp2 += 16; ap2 += 16;
    }

    float* pbase = part + (size_t)kc * B_ * H_;
    const int mbase = (lane >> 4) << 3;           // +8 rows for high half-wave
#pragma unroll
    for (int r = 0; r < 8; ++r) {
        pbase[(size_t)( 0 + r + mbase) * H_ + n] = acc0[r];
        pbase[(size_t)(16 + r + mbase) * H_ + n] = acc1[r];
        pbase[(size_t)(32 + r + mbase) * H_ + n] = acc2[r];
        pbase[(size_t)(48 + r + mbase) * H_ + n] = acc3[r];
    }
}

// ---------------------------------------------------------------------------
// Kernel 2b: mid = tanh(sum of 4 K-chunk partials + b_mid)
// ---------------------------------------------------------------------------
__global__ void __launch_bounds__(256)
mid_finish_kernel(const float* __restrict__ part, const float* __restrict__ bm,
                  float* __restrict__ mid) {
    const int i = blockIdx.x * 256 + threadIdx.x;   // 0 .. B*H-1 (65536, exact)
    const int n = i & (H_ - 1);
    const float s = part[i] + part[B_ * H_ + i] + part[2 * B_ * H_ + i]
                  + part[3 * B_ * H_ + i] + bm[n];
    mid[i] = tanhf(s);
}

// ---------------------------------------------------------------------------
// Kernel 3: logits = clip(mid @ W_vocab^T + b_vocab, +-50)   [64 x 50000]
// Written into the vocab_dist output region (overwritten later by finalize).
// Each W_vocab element is streamed from HBM exactly once (~205 MB -> ~8.8 us).
// ---------------------------------------------------------------------------
__global__ void __launch_bounds__(256)
gemm_vocab_kernel(const float* __restrict__ mid, const float* __restrict__ Wv,
                  const float* __restrict__ bv, float* __restrict__ logits) {
    const int lane  = threadIdx.x & 31;
    const int wave  = threadIdx.x >> 5;
    const int tileN = blockIdx.x * 8 + wave;      // 3125 valid tiles
    const int nloc  = lane & 15;
    const int n     = tileN * 16 + nloc;
    const int nc    = (n < V_) ? n : (V_ - 1);    // clamp loads; EXEC stays full
    const int kh    = (lane >> 4) << 1;

    v8f acc0 = {}, acc1 = {}, acc2 = {}, acc3 = {};
    const float* bp2 = Wv  + (size_t)nc * H_ + kh;   // W_vocab[n][k], K contiguous
    const float* ap2 = mid + (size_t)nloc * H_ + kh;

    for (int k0 = 0; k0 < H_; k0 += 16) {
#pragma unroll
        for (int u = 0; u < 16; u += 4) {
            const v2f bf = *(const v2f*)(bp2 + u);
            const v2f a0 = *(const v2f*)(ap2 + u);
            const v2f a1 = *(const v2f*)(ap2 + u + 16 * H_);
            const v2f a2 = *(const v2f*)(ap2 + u + 32 * H_);
            const v2f a3 = *(const v2f*)(ap2 + u + 48 * H_);
            acc0 = wmma_f32_k4(a0, bf, acc0);
            acc1 = wmma_f32_k4(a1, bf, acc1);
            acc2 = wmma_f32_k4(a2, bf, acc2);
            acc3 = wmma_f32_k4(a3, bf, acc3);
        }
        bp2 += 16; ap2 += 16;
    }

    if (n < V_) {
        const float bias  = bv[n];
        const int   mbase = (lane >> 4) << 3;
#pragma unroll
        for (int r = 0; r < 8; ++r) {
            float x0 = fminf(fmaxf(acc0[r] + bias, -50.f), 50.f);
            float x1 = fminf(fmaxf(acc1[r] + bias, -50.f), 50.f);
            float x2 = fminf(fmaxf(acc2[r] + bias, -50.f), 50.f);
            float x3 = fminf(fmaxf(acc3[r] + bias, -50.f), 50.f);
            logits[(size_t)( 0 + r + mbase) * V_ + n] = x0;
            logits[(size_t)(16 + r + mbase) * V_ + n] = x1;
            logits[(size_t)(32 + r + mbase) * V_ + n] = x2;
            logits[(size_t)(48 + r + mbase) * V_ + n] = x3;
        }
    }
}

// ---------------------------------------------------------------------------
// Kernel 4: per-row max and sum(exp(x - max)) over V, one block per row
// ---------------------------------------------------------------------------
__global__ void __launch_bounds__(256)
softmax_stats_kernel(const float* __restrict__ logits,
                     float* __restrict__ rmax, float* __restrict__ rsum) {
    __shared__ float red[256];
    const int b = blockIdx.x, t = threadIdx.x;
    float mx = -1e30f;
    for (int j = t; j < V_; j += 256)
        mx = fmaxf(mx, logits[(size_t)b * V_ + j]);
    red[t] = mx; __syncthreads();
    for (int off = 128; off > 0; off >>= 1) {
        if (t < off) red[t] = fmaxf(red[t], red[t + off]);
        __syncthreads();
    }
    const float m = red[0]; __syncthreads();
    float s = 0.f;
    for (int j = t; j < V_; j += 256)
        s += expf(logits[(size_t)b * V_ + j] - m);
    red[t] = s; __syncthreads();
    for (int off = 128; off > 0; off >>= 1) {
        if (t < off) red[t] += red[t + off];
        __syncthreads();
    }
    if (t == 0) { rmax[b] = m; rsum[b] = red[0]; }
}

// ---------------------------------------------------------------------------
// Kernel 5: vocab_dist = softmax (in place over logits); final = p_gen * dist,
//           zero the OOV tail of final_dist
// ---------------------------------------------------------------------------
__global__ void __launch_bounds__(256)
finalize_kernel(float* __restrict__ vout, float* __restrict__ fin,
                const float* __restrict__ rmax, const float* __restrict__ rsum,
                const float* __restrict__ pg) {
    const size_t t = (size_t)blockIdx.x * 256 + threadIdx.x;
    if (t >= (size_t)B_ * EXT_) return;
    const int b = (int)(t / EXT_);
    const int j = (int)(t % EXT_);
    if (j < V_) {
        const float l = vout[(size_t)b * V_ + j];
        const float e = expf(l - rmax[b]) / rsum[b];
        vout[(size_t)b * V_ + j] = e;
        fin[t] = pg[b] * e;
    } else {
        fin[t] = 0.f;
    }
}

// ---------------------------------------------------------------------------
// Kernel 6: copy distribution scatter-add: final[b, idx] += (1-p_gen)*attn
// ---------------------------------------------------------------------------
__global__ void __launch_bounds__(256)
scatter_kernel(const float* __restrict__ attn, const int* __restrict__ src_ids,
               const int* __restrict__ oov, const float* __restrict__ pg,
               float* __restrict__ fin) {
    const int t = blockIdx.x * 256 + threadIdx.x;
    if (t >= B_ * S_) return;
    const int b = t / S_;
    const int m = oov[t];
    const int idx = (m >= 0) ? (V_ + m) : src_ids[t];
    atomicAdd(&fin[(size_t)b * EXT_ + idx], (1.f - pg[b]) * attn[t]);
}

// ---------------------------------------------------------------------------
extern "C" void kernel_launch(void* const* d_in, const int* in_sizes, int n_in,
                              void* d_out, int out_size, void* d_ws, size_t ws_size,
                              hipStream_t stream) {
    const float* dec   = (const float*)d_in[0];   // [B,H]
    const float* ctx   = (const float*)d_in[1];   // [B,H]
    const float* emb   = (const float*)d_in[2];   // [B,E]
    const float* attn  = (const float*)d_in[3];   // [B,S]
    const int*   sids  = (const int*)  d_in[4];   // [B,S]
    const int*   soov  = (const int*)  d_in[5];   // [B,S]
    // d_in[6] = vocab_size (static V_)
    const float* Wm    = (const float*)d_in[7];   // [H,2H]
    const float* bm    = (const float*)d_in[8];   // [H]
    const float* Wp    = (const float*)d_in[9];   // [1,2H+E]
    const float* bp    = (const float*)d_in[10];  // [1]
    const float* Wv    = (const float*)d_in[11];  // [V,H]
    const float* bv    = (const float*)d_in[12];  // [V]

    // Output layout: final_dist [B,EXT] ++ p_gen [B] ++ vocab_dist [B,V]
    float* fin    = (float*)d_out;
    float* pg_out = fin + (size_t)B_ * EXT_;
    float* vout   = pg_out + B_;

    // Workspace: mid [B*H] ++ part [4*B*H] ++ pg [B] ++ rmax [B] ++ rsum [B]
    float* wsf  = (float*)d_ws;
    float* mid  = wsf;
    float* part = wsf + (size_t)B_ * H_;
    float* pg   = part + (size_t)4 * B_ * H_;
    float* rmax = pg + B_;
    float* rsum = rmax + B_;

    // 1) p_gen
    pgen_kernel<<<B_, 256, 0, stream>>>(dec, ctx, emb, Wp, bp, pg, pg_out);

    // 2a) mid partials: 4 K-chunks x 8 N-blocks = 32 blocks
    gemm_mid_partial_kernel<<<32, 256, 0, stream>>>(dec, ctx, Wm, part);

    // 2b) reduce partials + bias + tanh
    mid_finish_kernel<<<(B_ * H_) / 256, 256, 0, stream>>>(part, bm, mid);

    // 3) logits into vocab_dist output region (3125 tiles / 8 waves -> 391 blocks)
    gemm_vocab_kernel<<<(3125 + 7) / 8, 256, 0, stream>>>(mid, Wv, bv, vout);

    // 4) row max / sum(exp)
    softmax_stats_kernel<<<B_, 256, 0, stream>>>(vout, rmax, rsum);

    // 5) softmax in place + p_gen * dist + zero OOV tail
    finalize_kernel<<<(int)(((size_t)B_ * EXT_ + 255) / 256), 256, 0, stream>>>(
        vout, fin, rmax, rsum, pg);

    // 6) scatter-add copy distribution
    scatter_kernel<<<(B_ * S_ + 255) / 256, 256, 0, stream>>>(attn, sids, soov, pg, fin);
}